// SDGCN_31937376813495
// MI455X (gfx1250) — compile-verified
//
#include <hip/hip_runtime.h>

typedef __attribute__((ext_vector_type(2))) float v2f;
typedef __attribute__((ext_vector_type(8))) float v8f;

// ---------------------------------------------------------------- utilities

__global__ void zero_f32(float* __restrict__ p, int n) {
  int i = blockIdx.x * blockDim.x + threadIdx.x;
  if (i < n) p[i] = 0.0f;
}

__global__ void degree_kernel(const long long* __restrict__ row,
                              const long long* __restrict__ col,
                              float* __restrict__ deg_out,
                              float* __restrict__ deg_in, int E) {
  int e = blockIdx.x * blockDim.x + threadIdx.x;
  if (e < E) {
    unsafeAtomicAdd(&deg_out[(int)row[e]], 1.0f);
    unsafeAtomicAdd(&deg_in[(int)col[e]], 1.0f);
  }
}

__global__ void invsqrt_kernel(float* __restrict__ d, int n) {
  int i = blockIdx.x * blockDim.x + threadIdx.x;
  if (i < n) {
    float v = d[i];
    d[i] = (v > 0.0f) ? rsqrtf(fmaxf(v, 1.0f)) : 0.0f;
  }
}

__global__ void edgew_kernel(const long long* __restrict__ row,
                             const long long* __restrict__ col,
                             const float* __restrict__ inv_out,
                             const float* __restrict__ inv_in,
                             float* __restrict__ ew, int E) {
  int e = blockIdx.x * blockDim.x + threadIdx.x;
  if (e < E) ew[e] = inv_out[(int)row[e]] * inv_in[(int)col[e]];
}

// Wc = alpha*W + beta*Wd  (folds the direct term into one GEMM)
__global__ void combine_w(const float* __restrict__ W,
                          const float* __restrict__ Wd,
                          const float* __restrict__ alpha,
                          const float* __restrict__ beta,
                          float* __restrict__ Wc, int n) {
  int i = blockIdx.x * blockDim.x + threadIdx.x;
  if (i < n) Wc[i] = alpha[0] * W[i] + beta[0] * Wd[i];
}

__global__ void relu_kernel(float* __restrict__ p, long long n) {
  long long i = (long long)blockIdx.x * blockDim.x + threadIdx.x;
  if (i < n) p[i] = fmaxf(p[i], 0.0f);
}

// ------------------------------------------------- dual-output WMMA fp32 GEMM
// Computes T = X @ W and D = X @ Wc in one pass (A-fragments loaded once).
// X: [M x 128] row-major, W/Wc: [128 x NOUT] row-major.
// Block = 128 threads = 4 waves; each block owns a 16-row strip of X;
// wave w owns output columns [w*16*TPW, (w+1)*16*TPW).
template <int NOUT>
__global__ __launch_bounds__(128) void gemm_dual(const float* __restrict__ X,
                                                 const float* __restrict__ W,
                                                 const float* __restrict__ Wc,
                                                 float* __restrict__ T,
                                                 float* __restrict__ D) {
  constexpr int TPW = NOUT / 64;  // 16-wide column tiles per wave
  const int lane = threadIdx.x & 31;
  const int wave = threadIdx.x >> 5;
  const int half = lane >> 4;   // selects K pair (A) / K rows (B) / M half (C)
  const int l16  = lane & 15;   // M row (A) / N column (B, C/D)
  const int m0   = blockIdx.x * 16;

  const float* xrow = X + (size_t)(m0 + l16) * 128;

  v8f accT[TPW] = {};
  v8f accD[TPW] = {};

  for (int k = 0; k < 128; k += 4) {
    const int ka = k + 2 * half;
    // A fragment (16x4): lanes 0-15 hold K=k..k+1, lanes 16-31 hold K=k+2..k+3
    v2f a;
    a.x = xrow[ka];
    a.y = xrow[ka + 1];
#pragma unroll
    for (int t = 0; t < TPW; ++t) {
      const int c = wave * (16 * TPW) + t * 16 + l16;
      // B fragment (4x16): lane = N column, VGPR j / half-wave selects K row
      v2f bw, bc;
      bw.x = W [(size_t)ka * NOUT + c];
      bw.y = W [(size_t)(ka + 1) * NOUT + c];
      bc.x = Wc[(size_t)ka * NOUT + c];
      bc.y = Wc[(size_t)(ka + 1) * NOUT + c];
      accT[t] = __builtin_amdgcn_wmma_f32_16x16x4_f32(
          false, a, false, bw, (short)0, accT[t], false, false);
      accD[t] = __builtin_amdgcn_wmma_f32_16x16x4_f32(
          false, a, false, bc, (short)0, accD[t], false, false);
    }
  }

#pragma unroll
  for (int t = 0; t < TPW; ++t) {
    const int n0 = wave * (16 * TPW) + t * 16;
#pragma unroll
    for (int r = 0; r < 8; ++r) {
      const int m = m0 + r + 8 * half;  // C/D layout: VGPR r -> row r (+8 for upper half-wave)
      T[(size_t)m * NOUT + n0 + l16] = accT[t][r];
      D[(size_t)m * NOUT + n0 + l16] = accD[t][r];
    }
  }
}

// ---------------------------------------------- edge scatter: acc[row] += w*T[col]
template <int DIM>
__global__ void scatter_kernel(const long long* __restrict__ row,
                               const long long* __restrict__ col,
                               const float* __restrict__ ew,
                               const float* __restrict__ T,
                               float* __restrict__ acc, int E) {
  constexpr int PER = DIM / 4;  // float4 chunks per edge
  long long idx = (long long)blockIdx.x * blockDim.x + threadIdx.x;
  int e = (int)(idx / PER);
  if (e >= E) return;
  int j = (int)(idx % PER) * 4;
  float w = ew[e];
  int s = (int)col[e];
  int d = (int)row[e];
  const float4 v = *reinterpret_cast<const float4*>(T + (size_t)s * DIM + j);
  float* a = acc + (size_t)d * DIM + j;
  unsafeAtomicAdd(a + 0, w * v.x);
  unsafeAtomicAdd(a + 1, w * v.y);
  unsafeAtomicAdd(a + 2, w * v.z);
  unsafeAtomicAdd(a + 3, w * v.w);
}

// --------------------------------------------------------------------- launch

extern "C" void kernel_launch(void* const* d_in, const int* in_sizes, int n_in,
                              void* d_out, int out_size, void* d_ws, size_t ws_size,
                              hipStream_t stream) {
  const float*     x      = (const float*)d_in[0];
  const long long* ei     = (const long long*)d_in[1];
  const float*     W1     = (const float*)d_in[2];
  const float*     Wd1    = (const float*)d_in[3];
  const float*     alpha1 = (const float*)d_in[4];
  const float*     beta1  = (const float*)d_in[5];
  const float*     W2     = (const float*)d_in[6];
  const float*     Wd2    = (const float*)d_in[7];
  const float*     alpha2 = (const float*)d_in[8];
  const float*     beta2  = (const float*)d_in[9];
  float*           out    = (float*)d_out;

  const int N = in_sizes[0] / 128;   // 100000
  const int E = in_sizes[1] / 2;     // 1600000
  const long long* row = ei;
  const long long* col = ei + E;

  // workspace layout (floats)
  float* ws      = (float*)d_ws;
  float* deg     = ws;                         // 2N: deg_out | deg_in
  float* deg_out = ws;
  float* deg_in  = ws + N;
  float* ew      = ws + 2 * (size_t)N;         // E
  float* Wc1     = ew + E;                     // 128*128
  float* Wc2     = Wc1 + 128 * 128;            // 128*64
  float* T1      = Wc2 + 128 * 64;             // N*128
  float* H       = T1 + (size_t)N * 128;       // N*128
  float* T2      = H + (size_t)N * 128;        // N*64

  // normalization: D_out^{-1/2} A D_in^{-1/2} as per-edge weights
  zero_f32<<<(2 * N + 255) / 256, 256, 0, stream>>>(deg, 2 * N);
  degree_kernel<<<(E + 255) / 256, 256, 0, stream>>>(row, col, deg_out, deg_in, E);
  invsqrt_kernel<<<(2 * N + 255) / 256, 256, 0, stream>>>(deg, 2 * N);
  edgew_kernel<<<(E + 255) / 256, 256, 0, stream>>>(row, col, deg_out, deg_in, ew, E);

  // fold alpha/beta into combined weights
  combine_w<<<(128 * 128 + 255) / 256, 256, 0, stream>>>(W1, Wd1, alpha1, beta1, Wc1, 128 * 128);
  combine_w<<<(128 * 64 + 255) / 256, 256, 0, stream>>>(W2, Wd2, alpha2, beta2, Wc2, 128 * 64);

  // layer 1: T1 = x@W1, H = x@Wc1 (direct term), then H += scatter(w * T1[col])
  gemm_dual<128><<<N / 16, 128, 0, stream>>>(x, W1, Wc1, T1, H);
  {
    long long threads = (long long)E * 32;
    scatter_kernel<128><<<(unsigned)((threads + 255) / 256), 256, 0, stream>>>(row, col, ew, T1, H, E);
  }
  relu_kernel<<<(unsigned)(((long long)N * 128 + 255) / 256), 256, 0, stream>>>(H, (long long)N * 128);

  // layer 2: T2 = H@W2, out = H@Wc2, then out += scatter(w * T2[col])
  gemm_dual<64><<<N / 16, 128, 0, stream>>>(H, W2, Wc2, T2, out);
  {
    long long threads = (long long)E * 16;
    scatter_kernel<64><<<(unsigned)((threads + 255) / 256), 256, 0, stream>>>(row, col, ew, T2, out, E);
  }
}